// CRF_52201032516191
// MI455X (gfx1250) — compile-verified
//
#include <hip/hip_runtime.h>
#include <stdint.h>

#define BN 256
#define LN 1024
#define TN 64
#define NT 128              // threads per block: 4 waves on the WGP's 4 SIMD32s
#define CH 16               // timesteps per async-prefetch chunk
#define CHT (CH * TN)       // floats per chunk (1024 = 4 KB)
#define NSEG 64             // backtrack segments (16 steps each)
#define SEGL 16

// LDS layout (bytes):
//   [0,     512)    alpha ping-pong (2 * 64 f32)
//   [512,   1024)   partial best value from upper half (64 f32)
//   [1024,  1536)   partial best index (64 i32); reused as E[s] in backtrack
//   [1536,  9728)   pot double buffer (2 * CH * 64 f32); reused as G[64][64] u8
//   [9728,  13824)  tags (1024 i32)
//   [13824, 79360)  backpointers (1024 * 64 u8), bp[0] = identity
#define SMEM_BYTES 79360

// ---- CDNA5 async global->LDS (ASYNCcnt-tracked) ----------------------------
__device__ __forceinline__ void async_load_b128(const float* g, float* l) {
    unsigned lds_off = (unsigned)(uintptr_t)l;            // low 32 bits = LDS addr
    unsigned long long ga = (unsigned long long)(uintptr_t)g;
    asm volatile("global_load_async_to_lds_b128 %0, %1, off"
                 :: "v"(lds_off), "v"(ga) : "memory");
}
__device__ __forceinline__ void async_wait_le2() {
    asm volatile("s_wait_asynccnt 2" ::: "memory");       // prev chunk done
}
__device__ __forceinline__ void async_wait_0() {
    asm volatile("s_wait_asynccnt 0" ::: "memory");
}
// ---------------------------------------------------------------------------

__global__ __launch_bounds__(NT) void crf_viterbi_kernel(
    const float* __restrict__ pot,      // [B, L, T]
    const int*   __restrict__ lengths,  // [B]
    const float* __restrict__ trans,    // [T, T]
    int*         __restrict__ out)      // [B, L]
{
    extern __shared__ char smem[];
    float*         s_alpha = (float*)smem;                   // 2*TN
    float*         s_pv    = (float*)(smem + 512);           // TN
    int*           s_pi    = (int*)(smem + 1024);            // TN (also E[s])
    float*         s_pot   = (float*)(smem + 1536);          // 2*CHT (also G)
    int*           s_tags  = (int*)(smem + 9728);            // LN
    unsigned char* s_bp    = (unsigned char*)(smem + 13824); // LN*TN

    const int b   = blockIdx.x;
    const int tid = threadIdx.x;
    const int j   = tid & (TN - 1);     // tag owned by this thread
    const int h   = tid >> 6;           // predecessor half: i in [h*32, h*32+32)
    const int len = lengths[b];
    const float* potb = pot + (size_t)b * (LN * TN);

    // Half of transition column j in VGPRs (rows h*32 .. h*32+31).
    float tr[32];
#pragma unroll
    for (int i = 0; i < 32; ++i) tr[i] = trans[(h * 32 + i) * TN + j];

    // Async prefetch of chunk k (16 timesteps = 4 KB) into buffer k&1.
    auto issue_chunk = [&](int k) {
        const float* g = potb + (size_t)k * CHT;
        float* l = s_pot + (k & 1) * CHT;
#pragma unroll
        for (int q = 0; q < 2; ++q) {        // 2 b128 async loads per lane
            int f = q * 512 + tid * 4;       // coalesced 16B per lane
            async_load_b128(g + f, l + f);
        }
    };

    // Prologue: chunk0 + chunk1 in flight, wait for chunk0 (<=2 outstanding).
    issue_chunk(0);
    issue_chunk(1);
    async_wait_le2();
    __syncthreads();

    float alpha_j = 0.0f;                // live only in h==0 threads
    if (h == 0) {
        alpha_j = s_pot[j];              // alpha0 = pot[b,0,:]
        s_alpha[j] = alpha_j;
        s_bp[j] = (unsigned char)j;      // bp[0] = identity (pads segment 0)
    }
    __syncthreads();

    // ---------------- Forward max-plus DP ----------------
    int t = 1;
#pragma unroll 1
    for (int k = 0; k < LN / CH; ++k) {
        if (k > 0) {
            if (k + 1 < LN / CH) { issue_chunk(k + 1); async_wait_le2(); }
            else                 { async_wait_0(); }
            __syncthreads();             // chunk k visible to all waves
        }
        const float* prow = s_pot + (k & 1) * CHT;
        const int rstart = (k == 0) ? 1 : 0;
#pragma unroll 1
        for (int row = rstart; row < CH; ++row, ++t) {
            // Phase 1: each thread scans its 32 predecessors.
            const float* curA = s_alpha + ((t - 1) & 1) * TN + h * 32;
            float best = -3.402823466e38f;
            int bi = 0;
#pragma unroll
            for (int i = 0; i < 32; i += 4) {
                float4 av = *(const float4*)(curA + i);   // ds_load_b128 broadcast
                float s0 = av.x + tr[i + 0]; if (s0 > best) { best = s0; bi = i + 0; }
                float s1 = av.y + tr[i + 1]; if (s1 > best) { best = s1; bi = i + 1; }
                float s2 = av.z + tr[i + 2]; if (s2 > best) { best = s2; bi = i + 2; }
                float s3 = av.w + tr[i + 3]; if (s3 > best) { best = s3; bi = i + 3; }
            }
            if (h == 1) {                // upper half publishes its partial
                s_pv[j] = best;
                s_pi[j] = bi + 32;
            }
            __syncthreads();

            // Phase 2: lower-half thread combines (ties keep lower i => strict >).
            if (h == 0) {
                float v1 = s_pv[j];
                int   i1 = s_pi[j];
                if (v1 > best) { best = v1; bi = i1; }
                float potj = prow[row * TN + j];
                bool valid = (t < len);
                float na  = valid ? (best + potj) : alpha_j;  // freeze past length
                int   bpv = valid ? bi : j;                   // identity bp past length
                s_bp[t * TN + j] = (unsigned char)bpv;
                alpha_j = na;
                s_alpha[(t & 1) * TN + j] = na;
            }
            __syncthreads();
        }
    }

    // ---------------- Parallel segmented backtrack ----------------
    // F_t(x) = bp_t(x) maps tag_t -> tag_{t-1}; segment s composes t in
    // [16s, 16s+15]. G[s][x] = F_{16s}(...F_{16s+15}(x)...).
    unsigned char* s_G = (unsigned char*)s_pot;      // 64*64 u8 (pot is dead)

    // Phase A: thread (x=j, h) chases entry x through its 32 segments,
    // stepping all 32 independent chains together so the 32 ds_load_u8 per
    // iteration pipeline instead of serializing.
    {
        int cur[NSEG / 2];
#pragma unroll
        for (int ss = 0; ss < NSEG / 2; ++ss) cur[ss] = j;
#pragma unroll 1
        for (int step = 0; step < SEGL; ++step) {
#pragma unroll
            for (int ss = 0; ss < NSEG / 2; ++ss) {
                int s = h * (NSEG / 2) + ss;
                int tt = s * SEGL + (SEGL - 1) - step;
                cur[ss] = (int)s_bp[tt * TN + cur[ss]];
            }
        }
#pragma unroll
        for (int ss = 0; ss < NSEG / 2; ++ss) {
            int s = h * (NSEG / 2) + ss;
            s_G[s * TN + j] = (unsigned char)cur[ss];
        }
    }
    __syncthreads();

    // Phase B (serial, tiny): final argmax, then 64 boundary hops through G.
    if (tid == 0) {
        const float* fa = s_alpha + ((LN - 1) & 1) * TN;
        float bv = fa[0]; int bt = 0;
#pragma unroll 1
        for (int i = 1; i < TN; ++i) { if (fa[i] > bv) { bv = fa[i]; bt = i; } }
        int tag = bt;                                  // tag_{1023}
#pragma unroll 1
        for (int s = NSEG - 1; s >= 0; --s) {
            s_pi[s] = tag;                             // E[s] = tag_{16s+15}
            tag = (int)s_G[s * TN + tag];              // -> tag_{16s-1}
        }
    }
    __syncthreads();

    // Phase C: each of 64 threads replays its segment from E[s].
    if (h == 0) {
        int s = j;
        int tag = s_pi[s];
#pragma unroll
        for (int step = 0; step < SEGL; ++step) {
            int tt = s * SEGL + (SEGL - 1) - step;
            s_tags[tt] = tag;
            tag = (int)s_bp[tt * TN + tag];
        }
    }
    __syncthreads();

    // Coalesced output store: tags [B, L] int32.
#pragma unroll
    for (int q = 0; q < LN / NT; ++q) {
        int idx = q * NT + tid;
        out[(size_t)b * LN + idx] = s_tags[idx];
    }
}

extern "C" void kernel_launch(void* const* d_in, const int* in_sizes, int n_in,
                              void* d_out, int out_size, void* d_ws, size_t ws_size,
                              hipStream_t stream) {
    (void)in_sizes; (void)n_in; (void)out_size; (void)d_ws; (void)ws_size;
    const float* pot     = (const float*)d_in[0];   // [256,1024,64] f32
    const int*   lengths = (const int*)d_in[1];     // [256] i32
    const float* trans   = (const float*)d_in[2];   // [64,64] f32
    int*         out     = (int*)d_out;             // [256,1024] i32 tags

    // ~78 KB dynamic LDS per workgroup (CDNA5 WGP has 320 KB).
    hipFuncSetAttribute(reinterpret_cast<const void*>(crf_viterbi_kernel),
                        hipFuncAttributeMaxDynamicSharedMemorySize, SMEM_BYTES);
    crf_viterbi_kernel<<<BN, NT, SMEM_BYTES, stream>>>(pot, lengths, trans, out);
}